// NPTransitionPrior_56856777064611
// MI455X (gfx1250) — compile-verified
//
#include <hip/hip_runtime.h>
#include <math.h>

typedef __attribute__((ext_vector_type(2))) float v2f;
typedef __attribute__((ext_vector_type(8))) float v8f;
typedef __attribute__((ext_vector_type(4))) unsigned int v4u;
typedef __attribute__((ext_vector_type(8))) int v8i;
typedef __attribute__((ext_vector_type(4))) int v4i;

#define LAGS   2
#define SLOPE  0.01f
#define B_     64
#define T_     514
#define D_     32
#define H_     128
#define IN_    65
#define NROWS  (B_ * (T_ - LAGS))      /* 32768 */
#define RES_OFF  0
#define SUM_OFF  (NROWS * D_)          /* 1048576 */
#define HIST_OFF (SUM_OFF + B_)        /* 1048640 */

// One wave (32 threads) per block. blockIdx.x = 16-row tile of N (2048 tiles),
// blockIdx.y = feature d (32). W1[d] (128x65 f32) is staged to LDS via the
// Tensor Data Mover; the wave overlaps that with xx/yy/W2 staging, then runs
// both GEMMs on v_wmma_f32_16x16x4_f32.
__global__ __launch_bounds__(32)
void np_prior_main(const float* __restrict__ x,
                   const float* __restrict__ W1,
                   const float* __restrict__ b1,
                   const float* __restrict__ W2,
                   const float* __restrict__ b2,
                   float* __restrict__ out,
                   float* __restrict__ ws)
{
    __shared__ float w1s[H_ * IN_];   // W1[d][h][k], 33280 B (TDM target)
    __shared__ float xxs[16 * 64];    // A tile cols 0..63 (lags), 4 KB
    __shared__ float yys[16];         // yy = x[b, t+2, d]
    __shared__ float w2s[H_];
    __shared__ float gls[16 * H_];    // g tile, 8 KB

    const int lane  = threadIdx.x;        // 0..31
    const int half  = lane >> 4;          // 0/1
    const int lo    = lane & 15;          // row (A) or col (B/C) within 16
    const int rtile = blockIdx.x;         // 0..2047
    const int d     = blockIdx.y;         // 0..31
    const int n0    = rtile * 16;
    const int bidx  = n0 >> 9;            // batch element (512 rows per b)
    const int t0    = n0 & 511;

    // ---- TDM: DMA W1[d] (2D tile: 128 rows x 65 f32, row stride 65) to LDS
    {
        const unsigned long long ga =
            (unsigned long long)(const void*)(W1 + (size_t)d * (H_ * IN_));
        const unsigned int lds_addr =
            (unsigned int)(unsigned long long)(const void*)&w1s[0];

        // D# group 0: count=1 | lds_addr | global_addr[56:0] | type=2
        v4u g0;
        g0[0] = 1u;                                  // count=1 (valid descriptor)
        g0[1] = lds_addr;                            // LDS byte address
        g0[2] = (unsigned int)ga;                    // global_addr[31:0]
        g0[3] = (unsigned int)((ga >> 32) & 0x1FFFFFFu) | (2u << 30); // type=2

        // D# group 1: data_size=4B, dims/strides for 128x65 tile
        v8i g1;
        g1[0] = 0x00020000;          // workgroup_mask=0, data_size=2 (4 bytes)
        g1[1] = (IN_ << 16);         // tensor_dim0[15:0]=65  (bits 63:48)
        g1[2] = (H_ << 16);          // tensor_dim0[31:16]=0, tensor_dim1[15:0]=128
        g1[3] = (IN_ << 16);         // tensor_dim1[31:16]=0, tile_dim0=65
        g1[4] = H_;                  // tile_dim1=128, tile_dim2=0
        g1[5] = IN_;                 // tensor_dim0_stride[31:0]=65
        g1[6] = (int)((unsigned)(H_ * IN_) << 16); // dim1_stride[15:0]=8320
        g1[7] = 0;                   // dim1_stride[47:16]=0
        v4i gz4 = {0, 0, 0, 0};      // groups 2/3 unused (2D tensor)
        v8i gz8 = {0, 0, 0, 0, 0, 0, 0, 0};

        __builtin_amdgcn_tensor_load_to_lds(g0, g1, gz4, gz4, gz8, 0);
    }

    // ---- overlap with TDM: stage W2 row, xx tile, yy via normal loads ----
    {
        const float4* w2g4 = (const float4*)(W2 + d * H_);
        float4* w2s4 = (float4*)w2s;
        for (int i = lane; i < H_ / 4; i += 32) w2s4[i] = w2g4[i];
    }
    for (int r = 0; r < 16; ++r) {
        const float* xr = x + ((size_t)(bidx * T_ + t0 + r)) * D_;
        xxs[r * 64 + lane]      = xr[lane];        // l = 0
        xxs[r * 64 + 32 + lane] = xr[D_ + lane];   // l = 1
    }
    if (lane < 16) yys[lane] = x[((size_t)(bidx * T_ + t0 + lane + 2)) * D_ + d];

    __builtin_amdgcn_s_wait_tensorcnt(0);   // W1[d] tile resident in LDS
    __syncthreads();

    // ================= pre GEMM: (16x64) x (64x128) via wmma f32 16x16x4 ====
    v8f acc[8];
    #pragma unroll
    for (int hc = 0; hc < 8; ++hc)
        acc[hc] = (v8f){0.f, 0.f, 0.f, 0.f, 0.f, 0.f, 0.f, 0.f};

    #pragma unroll
    for (int hc = 0; hc < 8; ++hc) {
        const int hcol = hc * 16 + lo;         // B-fragment column (h)
        for (int kk = 0; kk < 16; ++kk) {      // K = 64 in steps of 4
            const int ka = kk * 4 + half * 2;  // A/B K index for this lane half
            v2f a, bb;
            a[0]  = xxs[lo * 64 + ka];
            a[1]  = xxs[lo * 64 + ka + 1];
            bb[0] = w1s[hcol * IN_ + ka];      // B[k][h] = W1[d][h][k]
            bb[1] = w1s[hcol * IN_ + ka + 1];
            acc[hc] = __builtin_amdgcn_wmma_f32_16x16x4_f32(
                false, a, false, bb, (short)0, acc[hc], false, false);
        }
    }

    // ===== rank-1 yy term + bias, activation, residual partials, g to LDS ===
    const float b2v = b2[d];
    float rpart[8];
    #pragma unroll
    for (int v = 0; v < 8; ++v) rpart[v] = 0.f;

    #pragma unroll
    for (int hc = 0; hc < 8; ++hc) {
        const int hcol = hc * 16 + lo;
        const float w1l = w1s[hcol * IN_ + 64];  // W1[d][h][64] (yy weight)
        const float b1v = b1[d * H_ + hcol];
        const float w2v = w2s[hcol];
        #pragma unroll
        for (int v = 0; v < 8; ++v) {
            const int m = half * 8 + v;                      // row in tile
            const float pre  = acc[hc][v] + yys[m] * w1l + b1v;
            const float actv = (pre >= 0.f) ? pre : SLOPE * pre;
            rpart[v] += actv * w2v;
            const float gv = ((pre >= 0.f) ? 1.f : SLOPE) * w2v;
            gls[m * H_ + hcol] = gv;
        }
    }

    // residuals: reduce over the 16 h-lanes of each half, lanes lo==0 write
    #pragma unroll
    for (int v = 0; v < 8; ++v) {
        float s = rpart[v];
        s += __shfl_xor(s, 1, 16);
        s += __shfl_xor(s, 2, 16);
        s += __shfl_xor(s, 4, 16);
        s += __shfl_xor(s, 8, 16);
        if (lo == 0)
            out[RES_OFF + (size_t)(n0 + half * 8 + v) * D_ + d] = s + b2v;
    }
    __syncthreads();

    // ================= jac GEMM: (16x128) x (128x65→80) via wmma ===========
    v8f jacc[5];
    #pragma unroll
    for (int kc = 0; kc < 5; ++kc)
        jacc[kc] = (v8f){0.f, 0.f, 0.f, 0.f, 0.f, 0.f, 0.f, 0.f};

    for (int hh = 0; hh < 32; ++hh) {          // K = 128 in steps of 4
        const int ha = hh * 4 + half * 2;
        v2f a;
        a[0] = gls[lo * H_ + ha];
        a[1] = gls[lo * H_ + ha + 1];
        #pragma unroll
        for (int kc = 0; kc < 5; ++kc) {
            const int col = kc * 16 + lo;      // output k column
            const bool ok = (col < IN_);
            v2f bb;
            bb[0] = ok ? w1s[ha * IN_ + col]       : 0.f;  // B2[h][k]=W1[d][h][k]
            bb[1] = ok ? w1s[(ha + 1) * IN_ + col] : 0.f;
            jacc[kc] = __builtin_amdgcn_wmma_f32_16x16x4_f32(
                false, a, false, bb, (short)0, jacc[kc], false, false);
        }
    }

    // hist_jac[d, n, 0, k] for k = 0..63
    float* hist = out + HIST_OFF + (size_t)d * ((size_t)NROWS * 64);
    #pragma unroll
    for (int kc = 0; kc < 4; ++kc) {
        #pragma unroll
        for (int v = 0; v < 8; ++v) {
            const int m = half * 8 + v;
            hist[(size_t)(n0 + m) * 64 + kc * 16 + lo] = jacc[kc][v];
        }
    }

    // log|jac[:, d, 64]| partial: column k=64 lives at kc=4, lo==0
    float p = 0.f;
    if (lo == 0) {
        #pragma unroll
        for (int v = 0; v < 8; ++v)
            p += logf(fabsf(jacc[4][v]));
    }
    p += __shfl_xor(p, 16);
    p += __shfl_xor(p, 8);
    p += __shfl_xor(p, 4);
    p += __shfl_xor(p, 2);
    p += __shfl_xor(p, 1);
    if (lane == 0) ws[(size_t)rtile * 32 + d] = p;
}

// Deterministic reduction of 1024 partials (32 row-tiles x 32 d) per batch b.
__global__ __launch_bounds__(32)
void np_prior_reduce(const float* __restrict__ ws, float* __restrict__ out)
{
    const int bb = blockIdx.x;     // 0..63
    const int lane = threadIdx.x;  // 0..31
    float s = 0.f;
    for (int i = lane; i < 1024; i += 32) s += ws[(size_t)bb * 1024 + i];
    s += __shfl_xor(s, 1);
    s += __shfl_xor(s, 2);
    s += __shfl_xor(s, 4);
    s += __shfl_xor(s, 8);
    s += __shfl_xor(s, 16);
    if (lane == 0) out[SUM_OFF + bb] = s;
}

extern "C" void kernel_launch(void* const* d_in, const int* in_sizes, int n_in,
                              void* d_out, int out_size, void* d_ws, size_t ws_size,
                              hipStream_t stream) {
    const float* x  = (const float*)d_in[0];
    const float* W1 = (const float*)d_in[1];
    const float* b1 = (const float*)d_in[2];
    const float* W2 = (const float*)d_in[3];
    const float* b2 = (const float*)d_in[4];
    float* out = (float*)d_out;
    float* ws  = (float*)d_ws;   // needs 2048*32 floats = 256 KB

    dim3 grid(NROWS / 16, D_);   // (2048, 32), one wave per block
    np_prior_main<<<grid, 32, 0, stream>>>(x, W1, b1, W2, b2, out, ws);
    np_prior_reduce<<<B_, 32, 0, stream>>>(ws, out);
}